// Encoder_56642028699704
// MI455X (gfx1250) — compile-verified
//
#include <hip/hip_runtime.h>

#define N_NODES_C  50000
#define N_EDGES_C  625000
#define FEAT       128
#define N_GRAPHS_C 128
#define N_CLASSES_C 10

typedef float v2f __attribute__((ext_vector_type(2)));
typedef float v8f __attribute__((ext_vector_type(8)));

// ---------------------------------------------------------------- utilities

__global__ void zero_kernel(float* __restrict__ p, int n) {
  int i = blockIdx.x * blockDim.x + threadIdx.x;
  if (i < n) p[i] = 0.0f;
}

// Pack a [K x N] row-major f32 weight matrix into V_WMMA_F32_16X16X4_F32
// B-fragment order:
//   pk[((kt*(N/16)+nt)*32 + lane)*2 + v] = W[(kt*4 + (lane>=16?2:0) + v)*N + nt*16 + (lane&15)]
// so each lane's B fragment is a contiguous float2 load.
__global__ void pack_b_kernel(const float* __restrict__ W, float* __restrict__ pk,
                              int K, int N) {
  int i = blockIdx.x * blockDim.x + threadIdx.x;
  if (i >= K * N) return;
  int v = i & 1;
  int t = i >> 1;
  int lane = t & 31; t >>= 5;
  int ntiles = N >> 4;
  int nt = t % ntiles;
  int kt = t / ntiles;
  int k = kt * 4 + ((lane >= 16) ? 2 : 0) + v;
  int n = nt * 16 + (lane & 15);
  pk[i] = W[(size_t)k * N + n];
}

// ------------------------------------------------- edge scatter (sum-aggregate)
// One wave per edge; lane l moves features [4l, 4l+4). float4 gather + 4 native
// f32 global atomics (no-return form) into agg[dst].
__global__ void scatter_add_kernel(const float* __restrict__ h,
                                   const int* __restrict__ src,
                                   const int* __restrict__ dst,
                                   float* __restrict__ agg) {
  long long t = (long long)blockIdx.x * blockDim.x + threadIdx.x;
  int e = (int)(t >> 5);
  if (e >= N_EDGES_C) return;
  int l = (int)(t & 31);
  int s = src[e];
  int d = dst[e];
  const float4 val = *(const float4*)(h + (size_t)s * FEAT + l * 4);
  float* p = agg + (size_t)d * FEAT + l * 4;
  unsafeAtomicAdd(p + 0, val.x);
  unsafeAtomicAdd(p + 1, val.y);
  unsafeAtomicAdd(p + 2, val.z);
  unsafeAtomicAdd(p + 3, val.w);
}

// ------------------------------------------------- fused GraphConv GEMM (WMMA)
// out = relu(agg @ Wr + h @ Ws + bias), all K = N = 128.
// One wave per 16-row M-tile; 8 x v8f accumulators cover N = 128.
// A-frag (16x4 f32): lane (m = m0 + lane%16) loads contiguous float2 at
// column kt*4 + (lane>=16 ? 2 : 0). B comes pre-packed fragment-major.
__global__ void __launch_bounds__(256)
gemm_layer_kernel(const float* __restrict__ agg, const float* __restrict__ h,
                  const float* __restrict__ pkWr, const float* __restrict__ pkWs,
                  const float* __restrict__ bias, float* __restrict__ out,
                  int nNodes, int mTiles) {
  int wave = threadIdx.x >> 5;
  int lane = threadIdx.x & 31;
  int mtile = blockIdx.x * 8 + wave;
  if (mtile >= mTiles) return;                 // wave-uniform: EXEC stays all-1s

  int mrow = mtile * 16 + (lane & 15);
  if (mrow > nNodes - 1) mrow = nNodes - 1;    // address clamp only (no EXEC change)
  int koff = (lane >= 16) ? 2 : 0;

  v8f acc[8];
#pragma unroll
  for (int nt = 0; nt < 8; ++nt) acc[nt] = (v8f)0.0f;

  const float* Asrc[2] = {agg, h};
  const float* Bsrc[2] = {pkWr, pkWs};

  for (int mat = 0; mat < 2; ++mat) {
    const float* arow = Asrc[mat] + (size_t)mrow * FEAT + koff;
    const float* Bp = Bsrc[mat];
#pragma unroll 4
    for (int kt = 0; kt < 32; ++kt) {          // K = 128, 4 per WMMA
      v2f a = *(const v2f*)(arow + kt * 4);
#pragma unroll
      for (int nt = 0; nt < 8; ++nt) {
        v2f b = *(const v2f*)(Bp + (size_t)(kt * 8 + nt) * 64 + lane * 2);
        acc[nt] = __builtin_amdgcn_wmma_f32_16x16x4_f32(
            false, a, false, b, (short)0, acc[nt], false, false);
      }
    }
  }

  // C/D layout: lane<16 -> M = r, lane>=16 -> M = 8 + r; N = nt*16 + lane%16
  int ncol = lane & 15;
  int rbase = mtile * 16 + ((lane >= 16) ? 8 : 0);
#pragma unroll
  for (int nt = 0; nt < 8; ++nt) {
    float bv = bias[nt * 16 + ncol];
#pragma unroll
    for (int r = 0; r < 8; ++r) {
      int row = rbase + r;
      if (row < nNodes) {
        float vv = acc[nt][r] + bv;
        out[(size_t)row * FEAT + nt * 16 + ncol] = vv > 0.0f ? vv : 0.0f;
      }
    }
  }
}

// ------------------------------------------------- per-graph max/mean pooling
__device__ __forceinline__ int lower_bound_i32(const int* __restrict__ a, int n, int key) {
  int lo = 0, hi = n;
  while (lo < hi) {
    int mid = (lo + hi) >> 1;
    if (a[mid] < key) lo = mid + 1; else hi = mid;
  }
  return lo;
}

// blockDim = 128 (one thread per feature), grid = N_GRAPHS. batch is sorted.
// g[gid][0:128] += max, g[gid][128:256] += mean.
__global__ void pool_accum_kernel(const float* __restrict__ h,
                                  const int* __restrict__ batch,
                                  float* __restrict__ g) {
  int gid = blockIdx.x;
  int f = threadIdx.x;
  int start = lower_bound_i32(batch, N_NODES_C, gid);
  int end = lower_bound_i32(batch, N_NODES_C, gid + 1);
  float mx = -3.402823466e+38f;
  float sm = 0.0f;
  for (int n = start; n < end; ++n) {
    float v = h[(size_t)n * FEAT + f];
    mx = fmaxf(mx, v);
    sm += v;
  }
  int cnt = end - start;
  float mean = (cnt > 0) ? sm / (float)cnt : 0.0f;
  if (cnt == 0) mx = 0.0f;                    // matches isfinite guard
  g[(size_t)gid * 256 + f] += mx;
  g[(size_t)gid * 256 + 128 + f] += mean;
}

// ------------------------------------------------- MLP head + log_softmax
// One block (128 threads) per graph; sizes 256->128->64->10.
__global__ void mlp_head_kernel(const float* __restrict__ g,
                                const float* __restrict__ L1w, const float* __restrict__ L1b,
                                const float* __restrict__ L2w, const float* __restrict__ L2b,
                                const float* __restrict__ L3w, const float* __restrict__ L3b,
                                float* __restrict__ out) {
  __shared__ float s_in[256];
  __shared__ float s_h1[128];
  __shared__ float s_h2[64];
  __shared__ float s_logit[N_CLASSES_C];
  int gi = blockIdx.x;
  int t = threadIdx.x;

  s_in[t] = g[(size_t)gi * 256 + t];
  s_in[t + 128] = g[(size_t)gi * 256 + 128 + t];
  __syncthreads();

  float acc = L1b[t];
  for (int k = 0; k < 256; ++k) acc += s_in[k] * L1w[(size_t)k * 128 + t];
  s_h1[t] = fmaxf(acc, 0.0f);
  __syncthreads();

  if (t < 64) {
    float a = L2b[t];
    for (int k = 0; k < 128; ++k) a += s_h1[k] * L2w[(size_t)k * 64 + t];
    s_h2[t] = fmaxf(a, 0.0f);
  }
  __syncthreads();

  if (t < N_CLASSES_C) {
    float a = L3b[t];
    for (int k = 0; k < 64; ++k) a += s_h2[k] * L3w[(size_t)k * N_CLASSES_C + t];
    s_logit[t] = a;
  }
  __syncthreads();

  if (t == 0) {
    float m = s_logit[0];
    for (int i = 1; i < N_CLASSES_C; ++i) m = fmaxf(m, s_logit[i]);
    float se = 0.0f;
    for (int i = 0; i < N_CLASSES_C; ++i) se += expf(s_logit[i] - m);
    float lse = m + logf(se);
    for (int i = 0; i < N_CLASSES_C; ++i)
      out[(size_t)gi * N_CLASSES_C + i] = s_logit[i] - lse;
  }
}

// ---------------------------------------------------------------- launcher

extern "C" void kernel_launch(void* const* d_in, const int* in_sizes, int n_in,
                              void* d_out, int out_size, void* d_ws, size_t ws_size,
                              hipStream_t stream) {
  const float* x    = (const float*)d_in[0];
  const int*   eidx = (const int*)d_in[1];
  const int*   src  = eidx;
  const int*   dst  = eidx + N_EDGES_C;
  const int*   batch = (const int*)d_in[2];
  const float* W1r = (const float*)d_in[3];
  const float* b1  = (const float*)d_in[4];
  const float* W1s = (const float*)d_in[5];
  const float* W2r = (const float*)d_in[6];
  const float* b2  = (const float*)d_in[7];
  const float* W2s = (const float*)d_in[8];
  const float* W3r = (const float*)d_in[9];
  const float* b3  = (const float*)d_in[10];
  const float* W3s = (const float*)d_in[11];
  const float* L1w = (const float*)d_in[12];
  const float* L1b = (const float*)d_in[13];
  const float* L2w = (const float*)d_in[14];
  const float* L2b = (const float*)d_in[15];
  const float* L3w = (const float*)d_in[16];
  const float* L3b = (const float*)d_in[17];
  float* out = (float*)d_out;

  const size_t nodeFeats = (size_t)N_NODES_C * FEAT;
  float* agg = (float*)d_ws;
  float* hA  = agg + nodeFeats;
  float* hB  = hA + nodeFeats;
  float* pk0 = hB + nodeFeats;            // 6 packed 128x128 weight matrices
  float* pk1 = pk0 + FEAT * FEAT;
  float* pk2 = pk1 + FEAT * FEAT;
  float* pk3 = pk2 + FEAT * FEAT;
  float* pk4 = pk3 + FEAT * FEAT;
  float* pk5 = pk4 + FEAT * FEAT;
  float* g   = pk5 + FEAT * FEAT;         // [128 x 256] pooled accumulator

  // Pack weights (B-fragment order for f32 WMMA).
  {
    int n = FEAT * FEAT;
    int blocks = (n + 255) / 256;
    pack_b_kernel<<<blocks, 256, 0, stream>>>(W1r, pk0, FEAT, FEAT);
    pack_b_kernel<<<blocks, 256, 0, stream>>>(W1s, pk1, FEAT, FEAT);
    pack_b_kernel<<<blocks, 256, 0, stream>>>(W2r, pk2, FEAT, FEAT);
    pack_b_kernel<<<blocks, 256, 0, stream>>>(W2s, pk3, FEAT, FEAT);
    pack_b_kernel<<<blocks, 256, 0, stream>>>(W3r, pk4, FEAT, FEAT);
    pack_b_kernel<<<blocks, 256, 0, stream>>>(W3s, pk5, FEAT, FEAT);
  }

  // Zero the pooled-feature accumulator.
  {
    int n = N_GRAPHS_C * 256;
    zero_kernel<<<(n + 255) / 256, 256, 0, stream>>>(g, n);
  }

  const int aggN = (int)nodeFeats;
  const int aggBlocks = (aggN + 255) / 256;
  const int scatterBlocks = (int)(((long long)N_EDGES_C * 32 + 255) / 256);
  const int mTiles = (N_NODES_C + 15) / 16;
  const int gemmBlocks = (mTiles + 7) / 8;

  // ---- Layer 1: h1 = relu(scatter(x) @ W1r + b1 + x @ W1s)
  zero_kernel<<<aggBlocks, 256, 0, stream>>>(agg, aggN);
  scatter_add_kernel<<<scatterBlocks, 256, 0, stream>>>(x, src, dst, agg);
  gemm_layer_kernel<<<gemmBlocks, 256, 0, stream>>>(agg, x, pk0, pk1, b1, hA,
                                                    N_NODES_C, mTiles);
  pool_accum_kernel<<<N_GRAPHS_C, 128, 0, stream>>>(hA, batch, g);

  // ---- Layer 2
  zero_kernel<<<aggBlocks, 256, 0, stream>>>(agg, aggN);
  scatter_add_kernel<<<scatterBlocks, 256, 0, stream>>>(hA, src, dst, agg);
  gemm_layer_kernel<<<gemmBlocks, 256, 0, stream>>>(agg, hA, pk2, pk3, b2, hB,
                                                    N_NODES_C, mTiles);
  pool_accum_kernel<<<N_GRAPHS_C, 128, 0, stream>>>(hB, batch, g);

  // ---- Layer 3
  zero_kernel<<<aggBlocks, 256, 0, stream>>>(agg, aggN);
  scatter_add_kernel<<<scatterBlocks, 256, 0, stream>>>(hB, src, dst, agg);
  gemm_layer_kernel<<<gemmBlocks, 256, 0, stream>>>(agg, hB, pk4, pk5, b3, hA,
                                                    N_NODES_C, mTiles);
  pool_accum_kernel<<<N_GRAPHS_C, 128, 0, stream>>>(hA, batch, g);

  // ---- MLP head + log_softmax
  mlp_head_kernel<<<N_GRAPHS_C, 128, 0, stream>>>(g, L1w, L1b, L2w, L2b,
                                                  L3w, L3b, out);
}